// SimpleNetwork_13056700579878
// MI455X (gfx1250) — compile-verified
//
#include <hip/hip_runtime.h>
#include <math.h>

typedef float v2f __attribute__((ext_vector_type(2)));
typedef float v8f __attribute__((ext_vector_type(8)));

#define NEG_SLOPE 0.01f
#define F_NODE 64
#define F_EDGE 8
#define MSG 4
#define EIN 136   // 2*F_NODE + F_EDGE
#define NIN 68    // F_NODE + MSG
#define N_GRAPHS 512
#define INTER 16

__device__ __forceinline__ void atomAddF(float* p, float v) {
    __hip_atomic_fetch_add(p, v, __ATOMIC_RELAXED, __HIP_MEMORY_SCOPE_AGENT);
}

// ---------------------------------------------------------------------------
// Edge message kernel: one thread per edge.
// m = softmax(leaky_relu([h[src], h[dst], e] @ fe_w.T + fe_b)); agg[src] += m
// ---------------------------------------------------------------------------
__global__ __launch_bounds__(256)
void edge_msg_kernel(const float* __restrict__ h,
                     const int* __restrict__ src,
                     const int* __restrict__ dst,
                     const float* __restrict__ eattr,
                     const float* __restrict__ few,   // [4,136]
                     const float* __restrict__ feb,   // [4]
                     float* __restrict__ agg,         // [N,4]
                     int E)
{
    __shared__ float Wsh[MSG * EIN + MSG];
    for (int i = threadIdx.x; i < MSG * EIN; i += blockDim.x) Wsh[i] = few[i];
    if (threadIdx.x < MSG) Wsh[MSG * EIN + threadIdx.x] = feb[threadIdx.x];
    __syncthreads();

    int e = blockIdx.x * blockDim.x + threadIdx.x;
    if (e >= E) return;

    int s = src[e];
    int d = dst[e];

    float acc[MSG];
#pragma unroll
    for (int k = 0; k < MSG; ++k) acc[k] = Wsh[MSG * EIN + k];

    const float4* hs = reinterpret_cast<const float4*>(h + (size_t)s * F_NODE);
#pragma unroll 4
    for (int j = 0; j < F_NODE / 4; ++j) {
        float4 a = hs[j];
#pragma unroll
        for (int k = 0; k < MSG; ++k) {
            const float* w = &Wsh[k * EIN + j * 4];
            acc[k] += a.x * w[0] + a.y * w[1] + a.z * w[2] + a.w * w[3];
        }
    }
    const float4* hd = reinterpret_cast<const float4*>(h + (size_t)d * F_NODE);
#pragma unroll 4
    for (int j = 0; j < F_NODE / 4; ++j) {
        float4 a = hd[j];
#pragma unroll
        for (int k = 0; k < MSG; ++k) {
            const float* w = &Wsh[k * EIN + F_NODE + j * 4];
            acc[k] += a.x * w[0] + a.y * w[1] + a.z * w[2] + a.w * w[3];
        }
    }
    const float4* ea = reinterpret_cast<const float4*>(eattr + (size_t)e * F_EDGE);
#pragma unroll
    for (int j = 0; j < F_EDGE / 4; ++j) {
        float4 a = ea[j];
#pragma unroll
        for (int k = 0; k < MSG; ++k) {
            const float* w = &Wsh[k * EIN + 2 * F_NODE + j * 4];
            acc[k] += a.x * w[0] + a.y * w[1] + a.z * w[2] + a.w * w[3];
        }
    }

    // leaky relu
#pragma unroll
    for (int k = 0; k < MSG; ++k) {
        float v = acc[k];
        acc[k] = v > 0.0f ? v : NEG_SLOPE * v;
    }
    // softmax over 4
    float mx = fmaxf(fmaxf(acc[0], acc[1]), fmaxf(acc[2], acc[3]));
    float ex[MSG];
    float sum = 0.0f;
#pragma unroll
    for (int k = 0; k < MSG; ++k) { ex[k] = __expf(acc[k] - mx); sum += ex[k]; }
    float inv = 1.0f / sum;

    float* ap = agg + (size_t)s * MSG;
#pragma unroll
    for (int k = 0; k < MSG; ++k) atomAddF(ap + k, ex[k] * inv);
}

// ---------------------------------------------------------------------------
// Node update via WMMA f32 16x16x4:  out = relu([h, agg] @ fh_w.T + fh_b)
// One wave handles a tile of 16 nodes x 64 outputs (4 accumulators).
// A frag: lane L -> row base+(L&15), float2 at col k0+2*(L>>4)  (ISA 16x4 f32 layout)
// B frag: lane L -> out col 16c+(L&15), float2 at fh_w[n][k0+2*(L>>4)]
// ---------------------------------------------------------------------------
__global__ __launch_bounds__(256)
void node_wmma_kernel(const float* __restrict__ h,     // [N,64]
                      const float* __restrict__ agg,   // [N,4]
                      const float* __restrict__ fhw,   // [64,68]
                      const float* __restrict__ fhb,   // [64]
                      float* __restrict__ out,         // [N,64]
                      int N, int ntile)
{
    int lane = threadIdx.x & 31;
    int wid  = blockIdx.x * (blockDim.x >> 5) + (threadIdx.x >> 5);
    if (wid >= ntile) return;              // uniform per wave -> EXEC stays all-ones

    int half = lane >> 4;                  // 0 or 1
    int base = wid * 16;
    int row  = base + (lane & 15);
    int rl   = row < N ? row : N - 1;      // clamp loads, keep EXEC full

    const float* hrow = h + (size_t)rl * F_NODE;
    const float* arow = agg + (size_t)rl * MSG;

    v8f acc[4];
#pragma unroll
    for (int c = 0; c < 4; ++c) acc[c] = (v8f){0.f,0.f,0.f,0.f,0.f,0.f,0.f,0.f};

    const float* wb[4];
#pragma unroll
    for (int c = 0; c < 4; ++c)
        wb[c] = fhw + (size_t)(c * 16 + (lane & 15)) * NIN;

    // K = 0..63 from h
#pragma unroll 4
    for (int k0 = 0; k0 < F_NODE; k0 += 4) {
        v2f a = *reinterpret_cast<const v2f*>(hrow + k0 + 2 * half);
#pragma unroll
        for (int c = 0; c < 4; ++c) {
            v2f b = *reinterpret_cast<const v2f*>(wb[c] + k0 + 2 * half);
            acc[c] = __builtin_amdgcn_wmma_f32_16x16x4_f32(
                false, a, false, b, (short)0, acc[c], false, false);
        }
    }
    // K = 64..67 from agg
    {
        v2f a = *reinterpret_cast<const v2f*>(arow + 2 * half);
#pragma unroll
        for (int c = 0; c < 4; ++c) {
            v2f b = *reinterpret_cast<const v2f*>(wb[c] + F_NODE + 2 * half);
            acc[c] = __builtin_amdgcn_wmma_f32_16x16x4_f32(
                false, a, false, b, (short)0, acc[c], false, false);
        }
    }

    // epilogue: bias + relu + store.  acc[c][j]: row = base + j + 8*half, col = 16c+(lane&15)
#pragma unroll
    for (int c = 0; c < 4; ++c) {
        int n = c * 16 + (lane & 15);
        float bias = fhb[n];
#pragma unroll
        for (int j = 0; j < 8; ++j) {
            int m = base + j + 8 * half;
            if (m < N) {
                float v = acc[c][j] + bias;
                out[(size_t)m * F_NODE + n] = fmaxf(v, 0.0f);
            }
        }
    }
}

// ---------------------------------------------------------------------------
// Per-graph scatter: sums[g] += h2[n], counts[g] += 1
// ---------------------------------------------------------------------------
__global__ __launch_bounds__(256)
void graph_scatter_kernel(const float* __restrict__ h2,
                          const int* __restrict__ batch,
                          float* __restrict__ sums,    // [512,64]
                          float* __restrict__ counts,  // [512]
                          int N)
{
    long long idx = (long long)blockIdx.x * blockDim.x + threadIdx.x;
    int n = (int)(idx >> 6);
    int f = (int)(idx & 63);
    if (n >= N) return;
    int g = batch[n];
    atomAddF(&sums[(size_t)g * F_NODE + f], h2[(size_t)n * F_NODE + f]);
    if (f == 0) atomAddF(&counts[g], 1.0f);
}

// ---------------------------------------------------------------------------
// Readout: mean -> relu(fc) -> relu(fz) -> softmax(2).  One thread per graph.
// ---------------------------------------------------------------------------
__global__ __launch_bounds__(256)
void readout_kernel(const float* __restrict__ sums,
                    const float* __restrict__ counts,
                    const float* __restrict__ fcw,  // [16,64]
                    const float* __restrict__ fcb,  // [16]
                    const float* __restrict__ fzw,  // [2,16]
                    const float* __restrict__ fzb,  // [2]
                    float* __restrict__ out)        // [512,2]
{
    int g = blockIdx.x * blockDim.x + threadIdx.x;
    if (g >= N_GRAPHS) return;
    float inv = 1.0f / fmaxf(counts[g], 1.0f);
    const float* srow = sums + (size_t)g * F_NODE;

    float inter[INTER];
#pragma unroll
    for (int i = 0; i < INTER; ++i) {
        float v = fcb[i];
        const float* w = fcw + (size_t)i * F_NODE;
        for (int j = 0; j < F_NODE; ++j) v += (srow[j] * inv) * w[j];
        inter[i] = fmaxf(v, 0.0f);
    }
    float z0 = fzb[0], z1 = fzb[1];
#pragma unroll
    for (int i = 0; i < INTER; ++i) {
        z0 += inter[i] * fzw[i];
        z1 += inter[i] * fzw[INTER + i];
    }
    z0 = fmaxf(z0, 0.0f);
    z1 = fmaxf(z1, 0.0f);
    float m = fmaxf(z0, z1);
    float e0 = __expf(z0 - m), e1 = __expf(z1 - m);
    float s = e0 + e1;
    out[(size_t)g * 2 + 0] = e0 / s;
    out[(size_t)g * 2 + 1] = e1 / s;
}

// ---------------------------------------------------------------------------
extern "C" void kernel_launch(void* const* d_in, const int* in_sizes, int n_in,
                              void* d_out, int out_size, void* d_ws, size_t ws_size,
                              hipStream_t stream)
{
    const float* x     = (const float*)d_in[0];
    const int*   eidx  = (const int*)d_in[1];
    const float* eattr = (const float*)d_in[2];
    const int*   batch = (const int*)d_in[3];
    const float* fe1w  = (const float*)d_in[4];
    const float* fe1b  = (const float*)d_in[5];
    const float* fh1w  = (const float*)d_in[6];
    const float* fh1b  = (const float*)d_in[7];
    const float* fe2w  = (const float*)d_in[8];
    const float* fe2b  = (const float*)d_in[9];
    const float* fh2w  = (const float*)d_in[10];
    const float* fh2b  = (const float*)d_in[11];
    const float* fcw   = (const float*)d_in[12];
    const float* fcb   = (const float*)d_in[13];
    const float* fzw   = (const float*)d_in[14];
    const float* fzb   = (const float*)d_in[15];

    const int N = in_sizes[0] / F_NODE;
    const int E = in_sizes[1] / 2;
    const int* src = eidx;
    const int* dst = eidx + E;

    float* agg    = (float*)d_ws;                    // N*4
    float* h1     = agg + (size_t)N * MSG;           // N*64
    float* h2     = h1 + (size_t)N * F_NODE;         // N*64
    float* sums   = h2 + (size_t)N * F_NODE;         // 512*64
    float* counts = sums + (size_t)N_GRAPHS * F_NODE;// 512

    const int ntile = (N + 15) / 16;
    const int node_blocks = (ntile * 32 + 255) / 256;
    const int edge_blocks = (E + 255) / 256;

    // ---- layer 1 ----
    hipMemsetAsync(agg, 0, (size_t)N * MSG * sizeof(float), stream);
    edge_msg_kernel<<<edge_blocks, 256, 0, stream>>>(x, src, dst, eattr, fe1w, fe1b, agg, E);
    node_wmma_kernel<<<node_blocks, 256, 0, stream>>>(x, agg, fh1w, fh1b, h1, N, ntile);

    // ---- layer 2 ----
    hipMemsetAsync(agg, 0, (size_t)N * MSG * sizeof(float), stream);
    edge_msg_kernel<<<edge_blocks, 256, 0, stream>>>(h1, src, dst, eattr, fe2w, fe2b, agg, E);
    node_wmma_kernel<<<node_blocks, 256, 0, stream>>>(h1, agg, fh2w, fh2b, h2, N, ntile);

    // ---- readout ----
    hipMemsetAsync(sums, 0, (size_t)(N_GRAPHS * F_NODE + N_GRAPHS) * sizeof(float), stream);
    long long scat_threads = (long long)N * F_NODE;
    int scat_blocks = (int)((scat_threads + 255) / 256);
    graph_scatter_kernel<<<scat_blocks, 256, 0, stream>>>(h2, batch, sums, counts, N);
    readout_kernel<<<(N_GRAPHS + 255) / 256, 256, 0, stream>>>(sums, counts, fcw, fcb, fzw, fzb, (float*)d_out);
}